// DOFENTransformer_83751862272391
// MI455X (gfx1250) — compile-verified
//
#include <hip/hip_runtime.h>
#include <hip/hip_bf16.h>

// ---------------------------------------------------------------------------
// Problem constants (from reference)
// ---------------------------------------------------------------------------
#define B_        16
#define N_COL     100
#define N_COND    64
#define D_        4
#define H_        128
#define N_RODT    1600            // N_COND*N_COL/D
#define N_SEQ     (B_ * N_RODT)   // 25600
#define N_FOREST  100
#define N_EST     256
#define N_CLASS   10
#define EPS_      1e-5f
#define SQRT_HD   11.313708498984761f   // sqrt(128): logits / (HEAD_DIM^-0.5)

typedef __attribute__((ext_vector_type(2))) float v2f;
typedef __attribute__((ext_vector_type(8))) float v8f;

// ---------------------------------------------------------------------------
// Wave helpers (wave32)
// ---------------------------------------------------------------------------
__device__ __forceinline__ float wave_sum(float v) {
#pragma unroll
  for (int m = 16; m > 0; m >>= 1) v += __shfl_xor(v, m, 32);
  return v;
}

__device__ __forceinline__ void softmax4(float* l) {
  float m = fmaxf(fmaxf(l[0], l[1]), fmaxf(l[2], l[3]));
  float s = 0.f;
#pragma unroll
  for (int i = 0; i < 4; ++i) { l[i] = __expf(l[i] - m); s += l[i]; }
  float inv = 1.0f / s;
#pragma unroll
  for (int i = 0; i < 4; ++i) l[i] *= inv;
}

// ---------------------------------------------------------------------------
// f32 WMMA GEMM:  C[M,N] = A[M,K] @ B[K,N] (+ colbias) (+ relu)
// One wave per 16-row x 128-col OUTPUT STRIP: 8 accumulator tiles share each
// A k-pair load, so one global_load_b64 of A feeds 8 V_WMMA_F32_16X16X4_F32.
// Requires: M % 16 == 0, N % 128 == 0, K % 4 == 0 (true for all uses here).
// epilogue: 0 = none, 1 = +bias, 2 = +bias then relu
// ---------------------------------------------------------------------------
__global__ void wmma_gemm_f32(const float* __restrict__ A,
                              const float* __restrict__ Bm,
                              const float* __restrict__ C_bias,
                              float* __restrict__ C,
                              int M, int N, int K, int epilogue) {
  int wave = (int)((blockIdx.x * blockDim.x + threadIdx.x) >> 5);
  int lane = (int)(threadIdx.x & 31);
  int ngrp  = N >> 7;                   // column groups of 128
  int total = (M >> 4) * ngrp;
  if (wave >= total) return;            // wave-uniform: EXEC stays all-ones
  int tm = wave / ngrp, ng = wave % ngrp;

  int half = lane >> 4;                 // 0: K pair {0,1}; 1: K pair {2,3}
  int l15  = lane & 15;
  int arow  = tm * 16 + l15;            // A row this lane feeds
  int cbase = ng * 128 + l15;           // B/C column for tile t is cbase + 16*t

  v8f acc[8];
#pragma unroll
  for (int t = 0; t < 8; ++t) acc[t] = (v8f){};

  for (int k0 = 0; k0 < K; k0 += 4) {
    int ka = k0 + half * 2;
    v2f a;                                        // contiguous -> one b64 load
    a[0] = A[(size_t)arow * K + ka];
    a[1] = A[(size_t)arow * K + ka + 1];
    const float* b0 = Bm + (size_t)ka * N + cbase;
    const float* b1 = Bm + (size_t)(ka + 1) * N + cbase;
#pragma unroll
    for (int t = 0; t < 8; ++t) {                 // A reused across 8 WMMAs
      v2f b;
      b[0] = b0[t * 16];
      b[1] = b1[t * 16];
      acc[t] = __builtin_amdgcn_wmma_f32_16x16x4_f32(
          /*neg_a=*/false, a, /*neg_b=*/false, b,
          /*c_mod=*/(short)0, acc[t], /*reuse_a=*/false, /*reuse_b=*/false);
    }
  }

  // D layout: VGPR r -> lanes 0-15: (M=tm*16+r, N=l15); lanes 16-31: M += 8
  int mbase = tm * 16 + half * 8;
#pragma unroll
  for (int t = 0; t < 8; ++t) {
    int cn = cbase + t * 16;
    float bv = (epilogue >= 1) ? C_bias[cn] : 0.0f;
#pragma unroll
    for (int r = 0; r < 8; ++r) {
      float v = acc[t][r] + bv;
      if (epilogue == 2) v = fmaxf(v, 0.0f);
      C[(size_t)(mbase + r) * N + cn] = v;
    }
  }
}

// ---------------------------------------------------------------------------
// Per-sequence kernel: one wave per rodt sequence (25600 waves).
// Builds q/k/v rows on the fly from rank-1 factorization, computes token-0
// attention for both w- and E-branches, residual + LayerNorm, w scalar,
// and stores E_tok for the WoE GEMM.
// Lane owns h-elements j = lane*4 + u, u in [0,4).
// ---------------------------------------------------------------------------
__global__ void rodt_kernel(const float* __restrict__ x,
                            const float* __restrict__ WnQ,   // 64 x 768
                            const float* __restrict__ BnQ,   // 64 x 768
                            const int*   __restrict__ perm,  // 100 x 64
                            const float* __restrict__ Wnum,  // 64 x 128
                            const float* __restrict__ bnum,  // 64 x 128
                            const float* __restrict__ gamma_w, const float* __restrict__ beta_w,
                            const float* __restrict__ gamma_E, const float* __restrict__ beta_E,
                            const float* __restrict__ Wow, const float* __restrict__ bow,
                            float* __restrict__ wbuf,        // 25600
                            float* __restrict__ Etok) {      // 25600 x 128
  int s = (int)((blockIdx.x * blockDim.x + threadIdx.x) >> 5);
  int lane = (int)(threadIdx.x & 31);
  if (s >= N_SEQ) return;

  int b = s / N_RODT, r = s % N_RODT;
  int c = r / 25, g = r % 25;

  int col0 = g * 4;
  float x0  = x[b * N_COL + col0];
  int   pc0 = perm[col0 * N_COND + c];
  const float* w0  = WnQ + (size_t)pc0 * 768;
  const float* q0b = BnQ + (size_t)pc0 * 768;

  float q0w[4], q0E[4], xa0[4];
#pragma unroll
  for (int u = 0; u < 4; ++u) {
    int j = lane * 4 + u;
    q0w[u] = fmaf(x0, w0[j],       q0b[j]);
    q0E[u] = fmaf(x0, w0[384 + j], q0b[384 + j]);
    xa0[u] = fmaf(x0, Wnum[(size_t)pc0 * H_ + j], bnum[(size_t)pc0 * H_ + j]);
  }

  float xd_[4]; int pcd_[4];
#pragma unroll
  for (int d = 0; d < 4; ++d) {
    int cold = g * 4 + d;
    xd_[d]  = x[b * N_COL + cold];
    pcd_[d] = perm[cold * N_COND + c];
  }

  // logits (row 0 only), both branches
  float lw[4], lE[4];
#pragma unroll
  for (int d = 0; d < 4; ++d) {
    const float* wr = WnQ + (size_t)pcd_[d] * 768;
    const float* br = BnQ + (size_t)pcd_[d] * 768;
    float xd = xd_[d];
    float pw = 0.f, pE = 0.f;
#pragma unroll
    for (int u = 0; u < 4; ++u) {
      int j = lane * 4 + u;
      float kw = fmaf(xd, wr[128 + j], br[128 + j]);
      float kE = fmaf(xd, wr[512 + j], br[512 + j]);
      pw = fmaf(q0w[u], kw, pw);
      pE = fmaf(q0E[u], kE, pE);
    }
    lw[d] = wave_sum(pw) * SQRT_HD;
    lE[d] = wave_sum(pE) * SQRT_HD;
  }
  softmax4(lw);
  softmax4(lE);

  // o0 = sum_d a_d * v_d  (recompute v rows to save registers)
  float ow[4] = {0, 0, 0, 0}, oE[4] = {0, 0, 0, 0};
#pragma unroll
  for (int d = 0; d < 4; ++d) {
    const float* wr = WnQ + (size_t)pcd_[d] * 768;
    const float* br = BnQ + (size_t)pcd_[d] * 768;
    float xd = xd_[d];
#pragma unroll
    for (int u = 0; u < 4; ++u) {
      int j = lane * 4 + u;
      float vw = fmaf(xd, wr[256 + j], br[256 + j]);
      float vE = fmaf(xd, wr[640 + j], br[640 + j]);
      ow[u] = fmaf(lw[d], vw, ow[u]);
      oE[u] = fmaf(lE[d], vE, oE[u]);
    }
  }

  // w branch: LN(ow + xa0) . Wow + bow
  {
    float t[4], ps = 0.f;
#pragma unroll
    for (int u = 0; u < 4; ++u) { t[u] = ow[u] + xa0[u]; ps += t[u]; }
    float mu = wave_sum(ps) * (1.0f / H_);
    float vs = 0.f;
#pragma unroll
    for (int u = 0; u < 4; ++u) { float dd = t[u] - mu; vs = fmaf(dd, dd, vs); }
    float rinv = rsqrtf(wave_sum(vs) * (1.0f / H_) + EPS_);
    float pdot = 0.f;
#pragma unroll
    for (int u = 0; u < 4; ++u) {
      int j = lane * 4 + u;
      float tok = fmaf((t[u] - mu) * rinv, gamma_w[j], beta_w[j]);
      pdot = fmaf(tok, Wow[j], pdot);
    }
    float tot = wave_sum(pdot);
    if (lane == 0) wbuf[s] = tot + bow[0];
  }

  // E branch: E_tok = LN(oE + xa0)
  {
    float t[4], ps = 0.f;
#pragma unroll
    for (int u = 0; u < 4; ++u) { t[u] = oE[u] + xa0[u]; ps += t[u]; }
    float mu = wave_sum(ps) * (1.0f / H_);
    float vs = 0.f;
#pragma unroll
    for (int u = 0; u < 4; ++u) { float dd = t[u] - mu; vs = fmaf(dd, dd, vs); }
    float rinv = rsqrtf(wave_sum(vs) * (1.0f / H_) + EPS_);
#pragma unroll
    for (int u = 0; u < 4; ++u) {
      int j = lane * 4 + u;
      Etok[(size_t)s * H_ + j] = fmaf((t[u] - mu) * rinv, gamma_E[j], beta_E[j]);
    }
  }
}

// ---------------------------------------------------------------------------
// Forest aggregation: one block (128 threads) per (b, forest).
// softmax over 256 gathered w values, weighted sum of gathered E rows,
// then LayerNorm(g1, be1). Output LNF (1600 x 128).
// ---------------------------------------------------------------------------
__global__ void forest_kernel(const float* __restrict__ wbuf,  // 25600
                              const float* __restrict__ E,     // 25600 x 128
                              const int*   __restrict__ swr,   // 100 x 256
                              const float* __restrict__ g1, const float* __restrict__ be1,
                              float* __restrict__ LNF) {       // 1600 x 128
  __shared__ float sw[N_EST];
  __shared__ int   sidx[N_EST];
  __shared__ float red[128];

  int bf = blockIdx.x;
  int b = bf / N_FOREST, f = bf % N_FOREST;
  int t = (int)threadIdx.x;

  for (int e = t; e < N_EST; e += 128) {
    int idx = swr[f * N_EST + e];
    sidx[e] = idx;
    sw[e] = wbuf[b * N_RODT + idx];
  }
  __syncthreads();

  // max over 256
  red[t] = fmaxf(sw[t], sw[t + 128]);
  __syncthreads();
  for (int st = 64; st > 0; st >>= 1) {
    if (t < st) red[t] = fmaxf(red[t], red[t + st]);
    __syncthreads();
  }
  float maxv = red[0];
  __syncthreads();

  // exp + sum
  float e0 = __expf(sw[t] - maxv), e1 = __expf(sw[t + 128] - maxv);
  sw[t] = e0; sw[t + 128] = e1;
  red[t] = e0 + e1;
  __syncthreads();
  for (int st = 64; st > 0; st >>= 1) {
    if (t < st) red[t] += red[t + st];
    __syncthreads();
  }
  float invs = 1.0f / red[0];
  __syncthreads();

  // weighted sum over 256 gathered E rows; thread t owns column t
  float acc = 0.f;
  for (int e = 0; e < N_EST; ++e) {
    if (e + 8 < N_EST)
      __builtin_prefetch(&E[(size_t)(b * N_RODT + sidx[e + 8]) * H_ + t], 0, 1);
    acc = fmaf(sw[e], E[(size_t)(b * N_RODT + sidx[e]) * H_ + t], acc);
  }
  acc *= invs;

  // LayerNorm across 128 columns
  red[t] = acc;
  __syncthreads();
  for (int st = 64; st > 0; st >>= 1) {
    if (t < st) red[t] += red[t + st];
    __syncthreads();
  }
  float mu = red[0] * (1.0f / H_);
  __syncthreads();
  float dd = acc - mu;
  red[t] = dd * dd;
  __syncthreads();
  for (int st = 64; st > 0; st >>= 1) {
    if (t < st) red[t] += red[t + st];
    __syncthreads();
  }
  float rinv = rsqrtf(red[0] * (1.0f / H_) + EPS_);
  LNF[(size_t)bf * H_ + t] = fmaf(dd * rinv, g1[t], be1[t]);
}

// ---------------------------------------------------------------------------
// Head: per row (b*100+f): LN(h1, g2, be2) @ W2 + b2, mean over forests.
// One wave per row; atomic accumulate 1/100 into out[b*10+cls].
// ---------------------------------------------------------------------------
__global__ void head_kernel(const float* __restrict__ h1,   // 1600 x 128
                            const float* __restrict__ g2, const float* __restrict__ be2,
                            const float* __restrict__ W2,   // 128 x 10
                            const float* __restrict__ b2,   // 10
                            float* __restrict__ out) {      // 16 x 10
  int s = (int)((blockIdx.x * blockDim.x + threadIdx.x) >> 5);
  int lane = (int)(threadIdx.x & 31);
  if (s >= B_ * N_FOREST) return;
  int b = s / N_FOREST;

  float t[4], ps = 0.f;
#pragma unroll
  for (int u = 0; u < 4; ++u) { t[u] = h1[(size_t)s * H_ + lane * 4 + u]; ps += t[u]; }
  float mu = wave_sum(ps) * (1.0f / H_);
  float vs = 0.f;
#pragma unroll
  for (int u = 0; u < 4; ++u) { float dd = t[u] - mu; vs = fmaf(dd, dd, vs); }
  float rinv = rsqrtf(wave_sum(vs) * (1.0f / H_) + EPS_);
  float ln[4];
#pragma unroll
  for (int u = 0; u < 4; ++u) {
    int j = lane * 4 + u;
    ln[u] = fmaf((t[u] - mu) * rinv, g2[j], be2[j]);
  }
#pragma unroll
  for (int cls = 0; cls < N_CLASS; ++cls) {
    float p = 0.f;
#pragma unroll
    for (int u = 0; u < 4; ++u) {
      int j = lane * 4 + u;
      p = fmaf(ln[u], W2[j * N_CLASS + cls], p);
    }
    float tot = wave_sum(p);
    if (lane == 0)
      atomicAdd(&out[b * N_CLASS + cls], (tot + b2[cls]) * (1.0f / N_FOREST));
  }
}

__global__ void zero_out_kernel(float* __restrict__ out, int n) {
  int i = (int)(blockIdx.x * blockDim.x + threadIdx.x);
  if (i < n) out[i] = 0.0f;
}

// ---------------------------------------------------------------------------
// Launch
// ---------------------------------------------------------------------------
extern "C" void kernel_launch(void* const* d_in, const int* in_sizes, int n_in,
                              void* d_out, int out_size, void* d_ws, size_t ws_size,
                              hipStream_t stream) {
  const float* x       = (const float*)d_in[0];
  // d_in[1] (quantile) unused by the reference
  const float* W_num   = (const float*)d_in[2];   // 64 x 128
  const float* b_num   = (const float*)d_in[3];   // 64 x 128
  const float* Wqkv    = (const float*)d_in[4];   // 128 x 768
  const float* bqkv    = (const float*)d_in[5];   // 768
  const float* Wow     = (const float*)d_in[6];   // 128
  const float* bow     = (const float*)d_in[7];   // 1
  const float* WoE     = (const float*)d_in[8];   // 128 x 128
  const float* boE     = (const float*)d_in[9];   // 128
  const float* gamma_w = (const float*)d_in[10];
  const float* beta_w  = (const float*)d_in[11];
  const float* gamma_E = (const float*)d_in[12];
  const float* beta_E  = (const float*)d_in[13];
  const float* g1      = (const float*)d_in[14];
  const float* be1     = (const float*)d_in[15];
  const float* W1      = (const float*)d_in[16];  // 128 x 128
  const float* b1      = (const float*)d_in[17];
  const float* g2      = (const float*)d_in[18];
  const float* be2     = (const float*)d_in[19];
  const float* W2      = (const float*)d_in[20];  // 128 x 10
  const float* b2      = (const float*)d_in[21];
  const int*   perm    = (const int*)d_in[22];    // 100 x 64
  const int*   swr     = (const int*)d_in[23];    // 100 x 256
  float* out = (float*)d_out;

  float* ws   = (float*)d_ws;
  float* WnQ  = ws; ws += 64 * 768;
  float* BnQ  = ws; ws += 64 * 768;
  float* wbuf = ws; ws += N_SEQ;
  float* Etok = ws; ws += (size_t)N_SEQ * H_;
  float* Ebuf = ws; ws += (size_t)N_SEQ * H_;
  float* LNF  = ws; ws += (size_t)B_ * N_FOREST * H_;
  float* h1   = ws;

  auto gemm = [&](const float* A, const float* Bm, const float* bias, float* C,
                  int M, int N, int K, int epi) {
    int waves  = (M / 16) * (N / 128);     // one wave per 16x128 strip
    int blocks = (waves * 32 + 255) / 256; // 8 waves per block
    wmma_gemm_f32<<<blocks, 256, 0, stream>>>(A, Bm, bias, C, M, N, K, epi);
  };

  // 1) Rank-1 precompute: WnQ = W_num @ Wqkv ; BnQ = b_num @ Wqkv + bqkv
  gemm(W_num, Wqkv, nullptr, WnQ, 64, 768, 128, 0);
  gemm(b_num, Wqkv, bqkv,    BnQ, 64, 768, 128, 1);

  // 2) Per-sequence attention (25600 waves)
  rodt_kernel<<<(N_SEQ * 32) / 256, 256, 0, stream>>>(
      x, WnQ, BnQ, perm, W_num, b_num, gamma_w, beta_w, gamma_E, beta_E,
      Wow, bow, wbuf, Etok);

  // 3) E = E_tok @ WoE + boE   (25600x128 @ 128x128)
  gemm(Etok, WoE, boE, Ebuf, N_SEQ, H_, H_, 1);

  // 4) Forest gather + softmax + weighted sum + LN  -> LNF (1600x128)
  forest_kernel<<<B_ * N_FOREST, 128, 0, stream>>>(wbuf, Ebuf, swr, g1, be1, LNF);

  // 5) h1 = relu(LNF @ W1 + b1)  (1600x128 @ 128x128)
  gemm(LNF, W1, b1, h1, B_ * N_FOREST, H_, H_, 2);

  // 6) head: zero output then LN(h1) @ W2 + b2, mean over forests
  zero_out_kernel<<<1, 256, 0, stream>>>(out, B_ * N_CLASS);
  head_kernel<<<(B_ * N_FOREST * 32 + 255) / 256, 256, 0, stream>>>(
      h1, g2, be2, W2, b2, out);
}